// HybridQuanvolutionNet_65481071410067
// MI455X (gfx1250) — compile-verified
//
#include <hip/hip_runtime.h>
#include <hip/hip_bf16.h>

// ---------------------------------------------------------------------------
// HybridQuanvolutionNet on MI455X (gfx1250, wave32).
// Kernel 1: one wave per 16-row tile of the (B*196, 16) psi matrix.
//   - build psi from 2x2 patch RY angles (fast sin/cos, products)
//   - re = psi @ U_re^T, im = psi @ U_im^T via 4 chained V_WMMA_F32_16X16X4_F32
//     each (fp32 end-to-end, matches reference precision)
//   - probs = re^2 + im^2 ; <Z_w> via signed butterfly (Walsh-Hadamard over N)
//   - fold into W_proj dot-product, wave-reduce, <=2 atomicAdd per wave
// Kernel 2: per-image head: -sin(proj+b), 10 logits, log_softmax.
// ---------------------------------------------------------------------------

typedef __attribute__((ext_vector_type(2))) float v2f;
typedef __attribute__((ext_vector_type(8))) float v8f;

#define BATCH 8192
#define ROWS (BATCH * 196)            // 1,605,632 = 16 * 100,352 (exact)
#define TILES (ROWS / 16)             // 100,352
#define WAVES_PER_BLOCK 8
#define BLOCKS_A (TILES / WAVES_PER_BLOCK)  // 12,544

__global__ __launch_bounds__(256) void quanv_wmma_kernel(
    const float* __restrict__ x,       // (B,1,28,28)
    const float* __restrict__ U_re,    // (16,16) row-major [n][k]
    const float* __restrict__ U_im,    // (16,16)
    const float* __restrict__ W_proj,  // (1,784)
    float* __restrict__ proj_acc)      // (B,) pre-zeroed accumulator
{
    const int wave = (blockIdx.x * 256 + threadIdx.x) >> 5;
    const int lane = threadIdx.x & 31;
    const int hi   = lane >> 4;        // which half of the wave
    const int l16  = lane & 15;
    const int row0 = wave << 4;        // first global (b,patch) row of tile

    // ---- per-lane patch for the A operand: matrix row m = l16 ----
    // (halves duplicate the psi computation; each half supplies K=0,1 vs 2,3)
    const int rowA = row0 + l16;
    const int bA = rowA / 196;
    const int pA = rowA % 196;
    const int prr = pA / 14, pcc = pA % 14;
    const float* px = x + (size_t)bA * 784 + prr * 56 + pcc * 2;
    const float2 q01 = *(const float2*)(px);       // x[r,c], x[r,c+1]
    const float2 q23 = *(const float2*)(px + 28);  // x[r+1,c], x[r+1,c+1]

    const float c0 = __cosf(0.5f * q01.x), s0 = __sinf(0.5f * q01.x);
    const float c1 = __cosf(0.5f * q01.y), s1 = __sinf(0.5f * q01.y);
    const float c2 = __cosf(0.5f * q23.x), s2 = __sinf(0.5f * q23.x);
    const float c3 = __cosf(0.5f * q23.y), s3 = __sinf(0.5f * q23.y);

    // psi[idx] = f0[i3]*f1[i2]*f2[i1]*f3[i0], idx = i3i2i1i0 (wire0 = MSB)
    float g[4], h[4], psi[16];
    g[0] = c0 * c1; g[1] = c0 * s1; g[2] = s0 * c1; g[3] = s0 * s1;
    h[0] = c2 * c3; h[1] = c2 * s3; h[2] = s2 * c3; h[3] = s2 * s3;
#pragma unroll
    for (int idx = 0; idx < 16; ++idx)
        psi[idx] = g[idx >> 2] * h[idx & 3];

    // ---- re/im = psi @ U^T via 4 chained V_WMMA_F32_16X16X4_F32 each ----
    // A (16x4 f32): lane<16 holds K = k0,k0+1 ; lane>=16 holds K = k0+2,k0+3
    // B (4x16 f32): same K split, N = l16; B[k][n] = U[n][k] (U row-major)
    v8f cre = {0.f, 0.f, 0.f, 0.f, 0.f, 0.f, 0.f, 0.f};
    v8f cim = {0.f, 0.f, 0.f, 0.f, 0.f, 0.f, 0.f, 0.f};
#pragma unroll
    for (int j = 0; j < 4; ++j) {
        const int k0 = 4 * j + 2 * hi;
        v2f a;
        a.x = psi[k0];
        a.y = psi[k0 + 1];
        const v2f br = *(const v2f*)(U_re + l16 * 16 + k0);
        const v2f bi = *(const v2f*)(U_im + l16 * 16 + k0);
        cre = __builtin_amdgcn_wmma_f32_16x16x4_f32(
            false, a, false, br, (short)0, cre, false, false);
        cim = __builtin_amdgcn_wmma_f32_16x16x4_f32(
            false, a, false, bi, (short)0, cim, false, false);
    }

    // probs: lane holds (m = r + 8*hi, n = l16) for r = 0..7
    v8f pr;
#pragma unroll
    for (int r = 0; r < 8; ++r)
        pr[r] = cre[r] * cre[r] + cim[r] * cim[r];

    // ---- <Z_w> via Walsh-Hadamard over the 16 N-lanes, fold into W_proj ----
    // After 4 signed-butterfly stages lane n holds sum_m p[m]*(-1)^popc(m&n);
    // n = 8,4,2,1 are exactly <Z_0..3>.
    const int bFirst = row0 / 196;
    const int bLast  = (row0 + 15) / 196;   // tile spans at most 2 images
    const bool active = (l16 == 1) | (l16 == 2) | (l16 == 4) | (l16 == 8);
    int w = 3;
    if (l16 == 2) w = 2; else if (l16 == 4) w = 1; else if (l16 == 8) w = 0;

    float acc0 = 0.f, acc1 = 0.f;
#pragma unroll
    for (int r = 0; r < 8; ++r) {
        float t = pr[r];
#pragma unroll
        for (int mask = 1; mask <= 8; mask <<= 1) {
            const float u = __shfl_xor(t, mask, 32);
            t = (l16 & mask) ? (u - t) : (t + u);
        }
        const int rowm = row0 + 8 * hi + r;
        const int bm = rowm / 196;
        const int pm = rowm % 196;
        float contrib = 0.f;
        if (active) contrib = t * W_proj[4 * pm + w];
        if (bm == bFirst) acc0 += contrib; else acc1 += contrib;
    }
#pragma unroll
    for (int mask = 16; mask >= 1; mask >>= 1) {
        acc0 += __shfl_xor(acc0, mask, 32);
        acc1 += __shfl_xor(acc1, mask, 32);
    }
    if (lane == 0) {
        atomicAdd(&proj_acc[bFirst], acc0);
        if (bLast != bFirst) atomicAdd(&proj_acc[bLast], acc1);
    }
}

__global__ __launch_bounds__(256) void head_kernel(
    const float* __restrict__ proj_acc,
    const float* __restrict__ b_proj,
    const float* __restrict__ W_cls,   // (10,1)
    const float* __restrict__ b_cls,   // (10,)
    float* __restrict__ out)           // (B,10)
{
    const int b = blockIdx.x * 256 + threadIdx.x;
    if (b >= BATCH) return;
    const float proj = proj_acc[b] + b_proj[0];
    const float qnn = -__sinf(proj);
    float lg[10];
    float mx = -3.402823466e38f;
#pragma unroll
    for (int j = 0; j < 10; ++j) {
        lg[j] = qnn * W_cls[j] + b_cls[j];
        mx = fmaxf(mx, lg[j]);
    }
    float se = 0.f;
#pragma unroll
    for (int j = 0; j < 10; ++j) se += __expf(lg[j] - mx);
    const float lse = __logf(se);
#pragma unroll
    for (int j = 0; j < 10; ++j) out[b * 10 + j] = lg[j] - mx - lse;
}

extern "C" void kernel_launch(void* const* d_in, const int* in_sizes, int n_in,
                              void* d_out, int out_size, void* d_ws, size_t ws_size,
                              hipStream_t stream) {
    const float* x      = (const float*)d_in[0];
    const float* U_re   = (const float*)d_in[1];
    const float* U_im   = (const float*)d_in[2];
    const float* W_proj = (const float*)d_in[3];
    const float* b_proj = (const float*)d_in[4];
    const float* W_cls  = (const float*)d_in[5];
    const float* b_cls  = (const float*)d_in[6];
    float* out = (float*)d_out;
    float* proj = (float*)d_ws;   // BATCH floats

    hipMemsetAsync(proj, 0, BATCH * sizeof(float), stream);
    quanv_wmma_kernel<<<BLOCKS_A, 256, 0, stream>>>(x, U_re, U_im, W_proj, proj);
    head_kernel<<<BATCH / 256, 256, 0, stream>>>(proj, b_proj, W_cls, b_cls, out);
}